// SigDotProductAttention_17093969838506
// MI455X (gfx1250) — compile-verified
//
#include <hip/hip_runtime.h>
#include <hip/hip_bf16.h>
#include <math.h>

// Problem constants (from reference): B=16, Tq=Tk=2048, D=512, fp32.
#define BATCH   16
#define SEQ     2048
#define DIM     512
#define QTILE   16          // q-rows per workgroup
#define NWAVES  8           // 256 threads
#define NTILES  (SEQ / 16)  // 128 key tiles of 16

typedef __attribute__((ext_vector_type(2))) float v2f;
typedef __attribute__((ext_vector_type(4))) float v4f;
typedef __attribute__((ext_vector_type(8))) float v8f;

// Fast sigmoid: v_exp_f32 + v_add + v_rcp_f32 (avoids the IEEE divide chain).
__device__ __forceinline__ float fast_sigmoid(float x) {
    return __builtin_amdgcn_rcpf(1.0f + __expf(-x));
}

// ---------------------------------------------------------------------------
// Kernel 1: one workgroup per (b, q-tile of 16 rows).
//  - stage Q tile in LDS (16x512 f32 = 32KB)
//  - each wave computes 16x16 score tiles with V_WMMA_F32_16X16X4_F32
//  - sigmoid -> LDS strip buf[16][2048] (128KB; 320KB WGP LDS makes this work)
//  - row-sum reduce in LDS, normalize, stream attn to global once (b128)
// ---------------------------------------------------------------------------
__global__ __launch_bounds__(256)
void sigattn_score_kernel(const float* __restrict__ Q,
                          const float* __restrict__ V,
                          float* __restrict__ attn)   // [B, Tq, Tk]
{
    __shared__ float qs[QTILE * DIM];        // 32 KB
    __shared__ float buf[QTILE * SEQ];       // 128 KB
    __shared__ float partial[256];
    __shared__ float invsum[QTILE];

    const int blk   = blockIdx.x;            // B * (Tq/16) = 2048 blocks
    const int b     = blk >> 7;              // / 128
    const int qt    = blk & 127;
    const int tid   = threadIdx.x;
    const int wave  = tid >> 5;
    const int lane  = tid & 31;

    const size_t q_base = ((size_t)b * SEQ + (size_t)qt * QTILE) * DIM;
    const size_t v_base = (size_t)b * SEQ * DIM;

    // ---- stage Q tile into LDS (vectorized: 2048 float4 / 256 threads) ----
    {
        const v4f* qg = (const v4f*)(Q + q_base);
        v4f*       qd = (v4f*)qs;
        #pragma unroll
        for (int i = tid; i < QTILE * DIM / 4; i += 256)
            qd[i] = qg[i];
    }
    __syncthreads();

    // ---- WMMA lane addressing (fp32 16x4 A / 4x16 B layout) ----
    // lane 0-15: M(or N)=lane, K pair {0,1}; lane 16-31: M=lane-16, K pair {2,3}
    const int mrow = lane & 15;
    const int koff = (lane >> 4) * 2;

    const float* qrow = qs + mrow * DIM + koff;                       // LDS
    // C/D layout: VGPR r, lanes 0-15 -> M=r ; lanes 16-31 -> M=r+8 ; N=lane&15
    const int crow0 = (lane >> 4) * 8;

    for (int nt = wave; nt < NTILES; nt += NWAVES) {
        const float* vrow = V + v_base + (size_t)(nt * 16 + mrow) * DIM + koff;

        v8f c = {};
        #pragma unroll 8
        for (int k0 = 0; k0 < DIM; k0 += 4) {
            v2f a  = *(const v2f*)(qrow + k0);   // ds_load_b64
            v2f bb = *(const v2f*)(vrow + k0);   // global_load_b64
            c = __builtin_amdgcn_wmma_f32_16x16x4_f32(
                    false, a, false, bb, (short)0, c, false, false);
        }

        // sigmoid + scatter into the LDS strip
        #pragma unroll
        for (int r = 0; r < 8; ++r) {
            buf[(r + crow0) * SEQ + nt * 16 + mrow] = fast_sigmoid(c[r]);
        }
    }
    __syncthreads();

    // ---- deterministic row sums: 16 threads/row, 128 keys each (b128) ----
    {
        const int row = tid >> 4;      // 0..15
        const int seg = tid & 15;      // 0..15
        const v4f* rp = (const v4f*)(buf + row * SEQ + seg * 128);
        v4f acc4 = {};
        #pragma unroll 8
        for (int i = 0; i < 32; ++i) acc4 += rp[i];
        partial[row * 16 + seg] = acc4.x + acc4.y + acc4.z + acc4.w;
    }
    __syncthreads();
    if (tid < QTILE) {
        float tot = 0.0f;
        #pragma unroll
        for (int s = 0; s < 16; ++s) tot += partial[tid * 16 + s];
        invsum[tid] = 1.0f / tot;      // 16 lanes once: precise divide is fine
    }
    __syncthreads();

    // ---- normalize + stream out, float4 (ds_load_b128 + global_store_b128) ----
    {
        v4f* out = (v4f*)(attn + ((size_t)b * SEQ + (size_t)qt * QTILE) * SEQ);
        const v4f* bp = (const v4f*)buf;
        #pragma unroll
        for (int v = tid; v < QTILE * SEQ / 4; v += 256) {
            int row = v >> 9;          // (v*4) / 2048
            out[v] = bp[v] * invsum[row];
        }
    }
}

// ---------------------------------------------------------------------------
// Kernel 2: w[b,k] = sum_q attn[b,q,k].  Deterministic (no atomics).
// Grid: B*8 blocks, 64 threads; each thread owns a float4 of key columns.
// Per-iteration each WG reads 1KB contiguous (fully coalesced b128).
// ---------------------------------------------------------------------------
__global__ __launch_bounds__(64)
void sigattn_colsum_kernel(const float* __restrict__ attn,
                           float* __restrict__ w)      // [B, Tk] in ws
{
    const int b  = blockIdx.x >> 3;
    const int kc = blockIdx.x & 7;                     // 8 chunks of 256 keys
    const int k4 = kc * 64 + threadIdx.x;              // float4 index within row

    const v4f* ap = (const v4f*)(attn + (size_t)b * SEQ * SEQ) + k4;
    v4f acc = {};
    #pragma unroll 4
    for (int q = 0; q < SEQ; ++q)
        acc += ap[(size_t)q * (SEQ / 4)];
    ((v4f*)(w + b * SEQ))[k4] = acc;
}

// ---------------------------------------------------------------------------
// Kernel 3: context[b,d] = sum_k w[b,k] * V[b,k,d].
// Grid: B blocks, 128 threads; each thread owns a float4 of d-columns.
// ---------------------------------------------------------------------------
__global__ __launch_bounds__(128)
void sigattn_context_kernel(const float* __restrict__ V,
                            const float* __restrict__ w,
                            float* __restrict__ ctx)   // [B, D]
{
    const int b  = blockIdx.x;
    const int d4 = threadIdx.x;                        // float4 index, 128 = 512/4

    const v4f* vb = (const v4f*)(V + (size_t)b * SEQ * DIM) + d4;
    const float* wb = w + b * SEQ;

    v4f acc = {};
    for (int k = 0; k < SEQ; ++k)
        acc += wb[k] * vb[(size_t)k * (DIM / 4)];
    ((v4f*)(ctx + b * DIM))[d4] = acc;
}

// ---------------------------------------------------------------------------
extern "C" void kernel_launch(void* const* d_in, const int* in_sizes, int n_in,
                              void* d_out, int out_size, void* d_ws, size_t ws_size,
                              hipStream_t stream) {
    const float* Q = (const float*)d_in[0];   // [16, 2048, 512]
    const float* V = (const float*)d_in[1];   // [16, 2048, 512]

    float* ctx  = (float*)d_out;                      // [16, 512]  (first output)
    float* attn = (float*)d_out + BATCH * DIM;        // [16, 2048, 2048] (second)
    float* w    = (float*)d_ws;                       // [16, 2048] colsum scratch

    (void)in_sizes; (void)n_in; (void)out_size; (void)ws_size;

    sigattn_score_kernel<<<BATCH * (SEQ / QTILE), 256, 0, stream>>>(Q, V, attn);
    sigattn_colsum_kernel<<<BATCH * 8, 64, 0, stream>>>(attn, w);
    sigattn_context_kernel<<<BATCH, 128, 0, stream>>>(V, w, ctx);
}